// ILIFNode_25838523253095
// MI455X (gfx1250) — compile-verified
//
#include <hip/hip_runtime.h>

// ILIF recurrence, T=16, N = B*C*H*W spatial lanes, float32 in/out.
// HBM-streaming-bound (512 MB total @ 23.3 TB/s => ~22 us floor).
// CDNA5 path: triple-buffered global->LDS async copies (ASYNCcnt) with
// non-temporal hints, ds_load_b128 consumption, NT b128 stores.

typedef float float4n __attribute__((ext_vector_type(4)));

#define T_STEPS 16
#define BLOCK   256
#define VEC     4
#define NBUF    3

__global__ __launch_bounds__(BLOCK)
void ilif_async_kernel(const float* __restrict__ x,
                       float* __restrict__ out,
                       long long n_per_step) {
    __shared__ float4n buf[NBUF][BLOCK];

    const int tid = threadIdx.x;
    const long long base =
        ((long long)blockIdx.x * BLOCK + (long long)tid) * VEC;

    const float* xptr = x + base;
    float*       optr = out + base;

    // Wave-relative LDS byte addresses of this lane's staging slots.
    // Generic (flat) shared pointers carry the LDS offset in addr[31:0].
    unsigned lds[NBUF];
    #pragma unroll
    for (int b = 0; b < NBUF; ++b)
        lds[b] = (unsigned)(unsigned long long)(&buf[b][tid]);

    // Prologue: put timesteps 0 and 1 in flight (streaming, bypass-friendly).
    asm volatile("global_load_async_to_lds_b128 %0, %1, off th:TH_LOAD_NT"
                 :: "v"(lds[0]), "v"(xptr) : "memory");
    asm volatile("global_load_async_to_lds_b128 %0, %1, off th:TH_LOAD_NT"
                 :: "v"(lds[1]), "v"(xptr + n_per_step) : "memory");

    float4n v = (float4n)0.0f;
    float4n s = (float4n)0.0f;

    #pragma unroll
    for (int t = 0; t < T_STEPS; ++t) {
        if (t + 2 < T_STEPS) {
            // Keep 2 loads in flight ahead of the consumer.
            const float* xnext = xptr + 2 * n_per_step;
            asm volatile("global_load_async_to_lds_b128 %0, %1, off th:TH_LOAD_NT"
                         :: "v"(lds[(t + 2) % NBUF]), "v"(xnext)
                         : "memory");
        }
        // Wait until the oldest outstanding copy (timestep t) has landed.
        const int outstanding_after_t = T_STEPS - 1 - t; // loads newer than t
        if (outstanding_after_t >= 2) {
            asm volatile("s_wait_asynccnt 2" ::: "memory");
        } else if (outstanding_after_t == 1) {
            asm volatile("s_wait_asynccnt 1" ::: "memory");
        } else {
            asm volatile("s_wait_asynccnt 0" ::: "memory");
        }

        // Consume this lane's staged tile (ds_load_b128).
        float4n xv = buf[t % NBUF][tid];

        // h = (v - s)*decay + x ; s = floor(clip(h,0,4)+0.5) ; v = h
        float4n h = (v - s) * 0.25f + xv;
        float4n sn;
        #pragma unroll
        for (int k = 0; k < VEC; ++k) {
            float c = fminf(fmaxf(h[k], 0.0f), 4.0f);  // -> v_med3_num_f32
            sn[k] = floorf(c + 0.5f);
        }
        v = h;
        s = sn;

        // Streaming output: nontemporal 128-bit store.
        __builtin_nontemporal_store(s, (float4n*)optr);

        xptr += n_per_step;
        optr += n_per_step;
    }
}

extern "C" void kernel_launch(void* const* d_in, const int* in_sizes, int n_in,
                              void* d_out, int out_size, void* d_ws, size_t ws_size,
                              hipStream_t stream) {
    (void)n_in; (void)d_ws; (void)ws_size; (void)out_size;

    const float* x = (const float*)d_in[0];
    float* out = (float*)d_out;

    const long long total = (long long)in_sizes[0];       // T * N
    const long long n_per_step = total / T_STEPS;         // N = B*C*H*W
    const long long threads = n_per_step / VEC;           // 4 floats/thread
    const int blocks = (int)((threads + BLOCK - 1) / BLOCK);

    ilif_async_kernel<<<blocks, BLOCK, 0, stream>>>(x, out, n_per_step);
}